// Sampler_21998822490203
// MI455X (gfx1250) — compile-verified
//
#include <hip/hip_runtime.h>
#include <hip/hip_bf16.h>
#include <math.h>

#define BDIM 512          // 16 wave32 waves per workgroup (one row per block)
#define NBINS 4096        // float-radix bins: sign+exp8+3 mantissa bits (u >> 20)
#define CAP 512           // candidate buffer capacity (expected ~80 for this dist)
#define VVOC 128000
#define VW (VVOC / 32)    // 4000 bitmask words
#define V4 (VVOC / 4)     // 32000 float4 per row

#if defined(__HIP_DEVICE_COMPILE__) && \
    __has_builtin(__builtin_amdgcn_global_load_async_to_lds_b128) && \
    __has_builtin(__builtin_amdgcn_s_wait_asynccnt)
#define USE_ASYNC 1
typedef int v4i __attribute__((vector_size(16)));          // <4 x i32>, 16B
typedef __attribute__((address_space(1))) v4i GV4;         // global int4
typedef __attribute__((address_space(3))) v4i LV4;         // LDS int4
#else
#define USE_ASYNC 0
#endif

__global__ __launch_bounds__(BDIM)
void Sampler_21998822490203_kernel(const float* __restrict__ logits,
                                   const int*   __restrict__ prev,
                                   const float* __restrict__ q,
                                   const float* __restrict__ temp_p,
                                   const int*   __restrict__ topk_p,
                                   const float* __restrict__ topp_p,
                                   const float* __restrict__ rp_p,
                                   int*         __restrict__ out,
                                   int Lprev)
{
    __shared__ unsigned int smask[VW];       // penalized-token bitmask (16000 B)
    __shared__ unsigned int hist[NBINS];     // 16384 B
    __shared__ float redF[BDIM];
    __shared__ float redS[BDIM];
    __shared__ int   redI[BDIM];
    __shared__ float candV[CAP];
    __shared__ int   candI[CAP];
    __shared__ float sortV[CAP];
    __shared__ int   sortI[CAP];
    __shared__ unsigned int candCnt;
    __shared__ unsigned int thrU;
    __shared__ float rowM, rowZ;
    __shared__ int   npKeep;
#if USE_ASYNC
    __shared__ __align__(16) float stage[2][BDIM * 4];   // 16384 B double buffer
#endif

    const int tid = threadIdx.x;
    const int row = blockIdx.x;

    // dynamic scalars (device-resident; loaded in-kernel, graph-capture safe)
    const float rp   = rp_p[0];
    const float topP = topp_p[0];
    float tv = temp_p[0];
    tv = (tv > 1e-5f) ? tv : 1e-5f;
    const float invT = 1.0f / tv;
    int K = topk_p[0];
    if (K < 1)   K = 1;
    if (K > CAP) K = CAP;

    // ---- init LDS ----
    for (int i = tid; i < VW; i += BDIM)    smask[i] = 0u;
    for (int i = tid; i < NBINS; i += BDIM) hist[i]  = 0u;
    if (tid == 0) { candCnt = 0u; npKeep = 0; }
    __syncthreads();

    // ---- build penalty bitmask from previous tokens ----
    const int* prow = prev + (size_t)row * Lprev;
    for (int i = tid; i < Lprev; i += BDIM) {
        unsigned int t = (unsigned int)prow[i];
        if (t < (unsigned int)VVOC) atomicOr(&smask[t >> 5], 1u << (t & 31));
    }
    __syncthreads();

    const float* lrow = logits + (size_t)row * VVOC;
    const float4* l4  = (const float4*)lrow;

    // per-thread online softmax state
    float m = -INFINITY, s = 0.0f;

    auto flipU = [](float v) -> unsigned int {
        unsigned int u = __float_as_uint(v);
        return (u & 0x80000000u) ? ~u : (u | 0x80000000u);
    };
    auto penal = [&](float v, unsigned int w, int sh) -> float {
        if ((w >> sh) & 1u) v = (v < 0.0f) ? v * rp : v / rp;
        return v;
    };
    auto acc1 = [&](float v) {
        if (v <= m) { s += expf(v - m); }
        else        { s = s * expf(m - v) + 1.0f; m = v; }
        atomicAdd(&hist[flipU(v) >> 20], 1u);
    };
    auto proc4 = [&](float4 vv, int t0) {
        unsigned int w = smask[t0 >> 5];   // t0 % 4 == 0 -> all 4 bits in one word
        int sh = t0 & 31;
        acc1(penal(vv.x, w, sh + 0));
        acc1(penal(vv.y, w, sh + 1));
        acc1(penal(vv.z, w, sh + 2));
        acc1(penal(vv.w, w, sh + 3));
    };

    // ---- pass 1: penalized online-softmax (M, Z) + float-radix histogram ----
    const int FI  = V4 / BDIM;            // 62 full iterations of float4
    const int REM = V4 - FI * BDIM;       // 256 tail float4s
#if USE_ASYNC
    // double-buffered async global->LDS staging (gfx1250 async-copy path)
    __builtin_amdgcn_global_load_async_to_lds_b128(
        (GV4*)(l4 + tid), (LV4*)(&stage[0][tid * 4]), 0, 0);
    for (int it = 0; it < FI; ++it) {
        const int buf = it & 1;
        if (it + 1 < FI) {
            __builtin_amdgcn_global_load_async_to_lds_b128(
                (GV4*)(l4 + (it + 1) * BDIM + tid),
                (LV4*)(&stage[buf ^ 1][tid * 4]), 0, 0);
            __builtin_amdgcn_s_wait_asynccnt(1);   // oldest copy complete
        } else {
            __builtin_amdgcn_s_wait_asynccnt(0);
        }
        float4 vv = *(const float4*)&stage[buf][tid * 4];
        proc4(vv, (it * BDIM + tid) * 4);
    }
#else
    for (int it = 0; it < FI; ++it) {
        int g = it * BDIM + tid;
        proc4(l4[g], g * 4);
    }
#endif
    if (tid < REM) {
        int g = FI * BDIM + tid;
        proc4(l4[g], g * 4);
    }

    // ---- reduce (m, s) -> row max M and full-vocab Z ----
    redF[tid] = m; redS[tid] = s;
    __syncthreads();
    for (int off = BDIM / 2; off > 0; off >>= 1) {
        if (tid < off) {
            float m1 = redF[tid], s1 = redS[tid];
            float m2 = redF[tid + off], s2 = redS[tid + off];
            float mm = fmaxf(m1, m2);
            redF[tid] = mm;
            redS[tid] = s1 * expf(m1 - mm) + s2 * expf(m2 - mm);
        }
        __syncthreads();
    }
    if (tid == 0) { rowM = redF[0]; rowZ = redS[0]; }

    // ---- find K-th value's bin: parallel suffix scan of histogram ----
    const int hi = NBINS - 1 - (tid << 3);   // each thread: 8 bins, descending
    unsigned int csum = 0;
    #pragma unroll
    for (int k = 0; k < 8; ++k) csum += hist[hi - k];
    redI[tid] = (int)csum;
    __syncthreads();
    for (int off = 1; off < BDIM; off <<= 1) {   // inclusive scan (desc order)
        int add = (tid >= off) ? redI[tid - off] : 0;
        __syncthreads();
        redI[tid] += add;
        __syncthreads();
    }
    {
        int incl = redI[tid];
        int excl = incl - (int)csum;
        if (excl < K && incl >= K) {             // exactly one thread wins
            unsigned int run = (unsigned int)excl;
            int b = hi;
            #pragma unroll
            for (int k = 0; k < 8; ++k) {
                run += hist[hi - k];
                if ((int)run >= K) { b = hi - k; break; }
            }
            thrU = ((unsigned int)b) << 20;      // lower bound of pivot bin
        }
    }
    __syncthreads();
    const unsigned int uThr = thrU;

    // ---- pass 2: collect candidates >= threshold bin (L2-resident reread) ----
    for (int g = tid; g < V4; g += BDIM) {
        float4 vv = l4[g];
        int t0 = g * 4;
        unsigned int w = smask[t0 >> 5];
        int sh = t0 & 31;
        float v0 = penal(vv.x, w, sh + 0);
        float v1 = penal(vv.y, w, sh + 1);
        float v2 = penal(vv.z, w, sh + 2);
        float v3 = penal(vv.w, w, sh + 3);
        if (flipU(v0) >= uThr) { unsigned p = atomicAdd(&candCnt, 1u); if (p < CAP) { candV[p] = v0; candI[p] = t0 + 0; } }
        if (flipU(v1) >= uThr) { unsigned p = atomicAdd(&candCnt, 1u); if (p < CAP) { candV[p] = v1; candI[p] = t0 + 1; } }
        if (flipU(v2) >= uThr) { unsigned p = atomicAdd(&candCnt, 1u); if (p < CAP) { candV[p] = v2; candI[p] = t0 + 2; } }
        if (flipU(v3) >= uThr) { unsigned p = atomicAdd(&candCnt, 1u); if (p < CAP) { candV[p] = v3; candI[p] = t0 + 3; } }
    }
    __syncthreads();
    const int n = (candCnt > CAP) ? CAP : (int)candCnt;   // n >= K by construction

    // ---- rank-sort candidates desc (stable: ties -> lower token index) ----
    for (int i = tid; i < n; i += BDIM) {
        float vi = candV[i]; int ii = candI[i];
        int r = 0;
        for (int j = 0; j < n; ++j) {
            float vj = candV[j]; int ij = candI[j];
            r += (vj > vi) || (vj == vi && ij < ii);
        }
        sortV[r] = vi; sortI[r] = ii;
    }
    __syncthreads();

    // ---- top-p: cum probs with FULL-vocab softmax denominator Z ----
    const float M = rowM;
    const float invZ = 1.0f / rowZ;
    float p = (tid < n) ? expf(sortV[tid] - M) * invZ : 0.0f;
    redS[tid] = p;
    __syncthreads();
    for (int off = 1; off < BDIM; off <<= 1) {   // inclusive prefix sum
        float add = (tid >= off) ? redS[tid - off] : 0.0f;
        __syncthreads();
        redS[tid] += add;
        __syncthreads();
    }
    const float cum = redS[tid];
    if (tid < n) {
        bool keep = (tid == 0) || (cum <= topP);  // removal is a suffix
        if (keep) atomicMax(&npKeep, tid + 1);
    }
    __syncthreads();
    const int np = npKeep;
    const float pivot = (np >= K) ? sortV[K - 1] : -INFINITY;

    // ---- survivors: argmax of v/T - log(q[idx])  (== argmax softmax/q) ----
    float score = -INFINITY;
    int   sidx  = 0x7fffffff;
    if (tid < np && sortV[tid] >= pivot) {
        int t = sortI[tid];
        score = sortV[tid] * invT - logf(q[(size_t)row * VVOC + t]);
        sidx  = t;
    }
    redS[tid] = score; redI[tid] = sidx;
    __syncthreads();
    for (int off = BDIM / 2; off > 0; off >>= 1) {
        if (tid < off) {
            float s2 = redS[tid + off]; int i2 = redI[tid + off];
            if (s2 > redS[tid] || (s2 == redS[tid] && i2 < redI[tid])) {
                redS[tid] = s2; redI[tid] = i2;
            }
        }
        __syncthreads();
    }
    if (tid == 0) out[row] = redI[0];
}

extern "C" void kernel_launch(void* const* d_in, const int* in_sizes, int n_in,
                              void* d_out, int out_size, void* d_ws, size_t ws_size,
                              hipStream_t stream) {
    const float* logits = (const float*)d_in[0];
    const int*   prev   = (const int*)  d_in[1];
    const float* q      = (const float*)d_in[2];
    const float* temp   = (const float*)d_in[3];
    const int*   topk   = (const int*)  d_in[4];
    const float* topp   = (const float*)d_in[5];
    const float* rp     = (const float*)d_in[6];
    const int B = in_sizes[0] / VVOC;
    const int L = in_sizes[1] / (B > 0 ? B : 1);
    Sampler_21998822490203_kernel<<<B, BDIM, 0, stream>>>(
        logits, prev, q, temp, topk, topp, rp, (int*)d_out, L);
}